// HierAttentionCopy_21552145891839
// MI455X (gfx1250) — compile-verified
//
#include <hip/hip_runtime.h>
#include <hip/hip_bf16.h>

// Problem constants (fixed by the reference's setup_inputs)
#define VOCAB   50000
#define B_      16
#define T_      32
#define NB_     8
#define L_      128
#define P_      (NB_ * L_)     // 1024 tokens per batch
#define CHUNK   12500          // vocab slots per LDS chunk (50,000 B LDS)
#define NCHUNK  4              // 4 * 12500 = 50000 = VOCAB
#define THREADS 256            // 8 waves (wave32)

// 128-bit payload type for the async LDS<->global DMA builtins
typedef int v4i_t __attribute__((ext_vector_type(4)));
typedef __attribute__((address_space(1))) v4i_t gv4i_t;  // global
typedef __attribute__((address_space(3))) v4i_t lv4i_t;  // LDS

__global__ __launch_bounds__(THREADS)
void hier_attn_copy_scatter(const float* __restrict__ bw,   // (B,T,NB)
                            const float* __restrict__ att,  // (B,T,NB,L)
                            const int*   __restrict__ idx,  // (B,NB,L) -> (B,P)
                            float* __restrict__ out)        // (B,T,VOCAB)
{
    __shared__ __align__(16) float acc[CHUNK];

    const int wg    = blockIdx.x;          // 0 .. B*T*NCHUNK-1
    const int chunk = wg & (NCHUNK - 1);
    const int bt    = wg >> 2;             // log2(NCHUNK) == 2
    const int b     = bt / T_;
    const int c0    = chunk * CHUNK;
    const int tid   = threadIdx.x;

    // ---- Phase 1: zero the LDS accumulator chunk (b128 stores) ----
    float4* acc4 = reinterpret_cast<float4*>(acc);
    #pragma unroll 4
    for (int i = tid; i < CHUNK / 4; i += THREADS)
        acc4[i] = make_float4(0.f, 0.f, 0.f, 0.f);
    __syncthreads();

    // ---- Phase 2: scatter-add x = att * block_weight into LDS by vocab id ----
    // ds_add_f32 atomics accumulate duplicate token ids correctly.
    for (int p = tid; p < P_; p += THREADS) {
        const int   id  = idx[b * P_ + p];
        const int   nb  = p >> 7;                        // p / L_
        const float x   = att[(size_t)bt * P_ + p] * bw[bt * NB_ + nb];
        const int   rel = id - c0;
        if ((unsigned)rel < (unsigned)CHUNK)
            atomicAdd(&acc[rel], x);                     // LDS float atomic
    }
    __syncthreads();

    // ---- Phase 3: stream LDS chunk -> global output slice ----
    float* dst = out + (size_t)bt * VOCAB + c0;          // 16B aligned
#if __has_builtin(__builtin_amdgcn_global_store_async_from_lds_b128)
    // CDNA5 async LDS->memory DMA path (VGPR-bypassing, tracked by ASYNCcnt)
    for (int i = tid; i < CHUNK / 4; i += THREADS) {
        __builtin_amdgcn_global_store_async_from_lds_b128(
            (gv4i_t*)(dst + 4 * i),
            (lv4i_t*)(acc + 4 * i),
            /*offset=*/0, /*cpol=*/0);
    }
  #if __has_builtin(__builtin_amdgcn_s_wait_asynccnt)
    __builtin_amdgcn_s_wait_asynccnt(0);
  #else
    asm volatile("s_wait_asynccnt 0" ::: "memory");
  #endif
#else
    // Fallback: plain b128 copies through VGPRs
    float4* dst4 = reinterpret_cast<float4*>(dst);
    for (int i = tid; i < CHUNK / 4; i += THREADS)
        dst4[i] = acc4[i];
#endif
}

extern "C" void kernel_launch(void* const* d_in, const int* in_sizes, int n_in,
                              void* d_out, int out_size, void* d_ws, size_t ws_size,
                              hipStream_t stream) {
    const float* bw  = (const float*)d_in[0];   // block_weight (B,T,NB)
    const float* att = (const float*)d_in[1];   // att          (B,T,NB,L)
    const int*   idw = (const int*)  d_in[2];   // in_word      (B,NB,L)
    float*       out = (float*)d_out;           // (B,T,VOCAB)

    dim3 grid(B_ * T_ * NCHUNK);                // 2048 workgroups
    dim3 block(THREADS);
    hier_attn_copy_scatter<<<grid, block, 0, stream>>>(bw, att, idw, out);
}